// PerParticleTensorPredictor_19464791785667
// MI455X (gfx1250) — compile-verified
//
#include <hip/hip_runtime.h>
#include <hip/hip_bf16.h>
#include <math.h>

typedef __attribute__((ext_vector_type(2))) float v2f;
typedef __attribute__((ext_vector_type(8))) float v8f;

#define N_TOT        65536
#define F            128
#define ROWS_PER_N   18          // P(2) * LM(9)
#define N_PER_WAVE   8           // 8 n -> 144 rows -> 9 tiles of 16
#define TILES_PER_WAVE 9
#define WAVES_PER_WG 2
#define WG_SIZE      64

// workspace layout (float offsets)
#define WS_U     0      // 2048: U[k*16+c], c = t*6+p*3+l (c<12), else 0
#define WS_CG    2048   // 729 : real-basis CG tensor (9x9x9)
#define WS_WCG   2777   // 729 : CG * wpath gather
#define WS_CG112 3506   // 81  : CG[1:4,1:4,:] as [(a*3+b)*9+l]
#define WS_CB    3587   // 2   : bias constants for x1/x2

__device__ __constant__ int lof9[9] = {0,1,1,1,2,2,2,2,2};

// ---------------- precompute: CG tensor + collapsed weights ----------------

__device__ double dfact(int n){ double r=1.0; for(int i=2;i<=n;++i) r*=(double)i; return r; }

__device__ double cg_complex(int j1,int m1,int j2,int m2,int j3,int m3){
  int lo = (j1>j2)? j1-j2 : j2-j1;
  if (m3 != m1+m2 || j3 < lo || j3 > j1+j2) return 0.0;
  double pref = sqrt((double)(2*j3+1) * dfact(j1+j2-j3)*dfact(j1-j2+j3)*dfact(-j1+j2+j3)/dfact(j1+j2+j3+1));
  pref *= sqrt(dfact(j1+m1)*dfact(j1-m1)*dfact(j2+m2)*dfact(j2-m2)*dfact(j3+m3)*dfact(j3-m3));
  double s = 0.0;
  for (int k=0;k<=j1+j2-j3;++k){
    int t2=j1-m1-k, t3=j2+m2-k, t4=j3-j2+m1+k, t5=j3-j1-m2+k;
    if (t2<0||t3<0||t4<0||t5<0) continue;
    double d = dfact(k)*dfact(j1+j2-j3-k)*dfact(t2)*dfact(t3)*dfact(t4)*dfact(t5);
    s += ((k&1)? -1.0:1.0)/d;
  }
  return pref*s;
}

__device__ void umat(int l, double Ur[5][5], double Ui[5][5]){
  for(int a=0;a<5;a++) for(int b=0;b<5;b++){ Ur[a][b]=0.0; Ui[a][b]=0.0; }
  Ur[l][l]=1.0;
  const double is2 = 0.70710678118654752440;
  for(int m=1;m<=l;m++){
    double sgn = (m&1)? -1.0 : 1.0;
    Ur[l+m][l-m] = is2;
    Ur[l+m][l+m] = sgn*is2;
    Ui[l-m][l-m] = is2;
    Ui[l-m][l+m] = -sgn*is2;
  }
}

__global__ __launch_bounds__(256)
void pptp_precompute(const float* __restrict__ W3,  const float* __restrict__ b3,
                     const float* __restrict__ Wt1, const float* __restrict__ bt1,
                     const float* __restrict__ Wt2, const float* __restrict__ bt2,
                     const float* __restrict__ wpath, float* __restrict__ ws)
{
  const int tid = threadIdx.x;

  // U[k*16+c] = sum_o W3[p,l,k,o] * Wt{t}[p,l,o,0]
  for (int idx=tid; idx<2048; idx+=256){
    int k = idx>>4, c = idx&15;
    float sum = 0.f;
    if (c < 12){
      int t=c/6, r=c%6, p=r/3, l=r%3;
      const float* W  = W3 + (size_t)((p*3+l)*128 + k)*128;
      const float* wt = (t==0 ? Wt1 : Wt2) + (p*3+l)*128;
      for (int o=0;o<128;++o) sum += W[o]*wt[o];
    }
    ws[WS_U+idx] = sum;
  }
  for (int i=tid;i<729;i+=256) ws[WS_CG+i]=0.f;
  if (tid==0){
    float s1=0.f, s2=0.f;
    for(int o=0;o<128;++o){ s1 += b3[o]*Wt1[o]; s2 += b3[o]*Wt2[o]; }
    ws[WS_CB+0]=s1+bt1[0]; ws[WS_CB+1]=s2+bt2[0];
  }
  __syncthreads();

  if (tid < 27){
    int l1=tid/9, l2=(tid/3)%3, l3=tid%3;
    int lo = (l1>l2)? l1-l2 : l2-l1;
    if (l3>=lo && l3<=l1+l2){
      int d1=2*l1+1, d2=2*l2+1, d3=2*l3+1;
      double C[5][5][5];
      for(int a=0;a<5;a++)for(int b=0;b<5;b++)for(int c=0;c<5;c++) C[a][b][c]=0.0;
      for(int m1=-l1;m1<=l1;m1++)
        for(int m2=-l2;m2<=l2;m2++){
          int m3=m1+m2;
          if (m3>=-l3 && m3<=l3) C[l1+m1][l2+m2][l3+m3]=cg_complex(l1,m1,l2,m2,l3,m3);
        }
      double U1r[5][5],U1i[5][5],U2r[5][5],U2i[5][5],U3r[5][5],U3i[5][5];
      umat(l1,U1r,U1i); umat(l2,U2r,U2i); umat(l3,U3r,U3i);
      double A1r[5][5][5],A1i[5][5][5],A2r[5][5][5],A2i[5][5][5];
      for(int a=0;a<d1;a++)for(int n=0;n<d2;n++)for(int o=0;o<d3;o++){
        double sr=0,si=0;
        for(int m=0;m<d1;m++){ double cv=C[m][n][o]; sr+=U1r[a][m]*cv; si+=-U1i[a][m]*cv; }
        A1r[a][n][o]=sr; A1i[a][n][o]=si;
      }
      for(int a=0;a<d1;a++)for(int b=0;b<d2;b++)for(int o=0;o<d3;o++){
        double sr=0,si=0;
        for(int n=0;n<d2;n++){
          double cr=U2r[b][n], ci=-U2i[b][n];
          sr += cr*A1r[a][n][o]-ci*A1i[a][n][o];
          si += cr*A1i[a][n][o]+ci*A1r[a][n][o];
        }
        A2r[a][b][o]=sr; A2i[a][b][o]=si;
      }
      int par=(l1+l2+l3)&1;
      for(int a=0;a<d1;a++)for(int b=0;b<d2;b++)for(int c=0;c<d3;c++){
        double sr=0,si=0;
        for(int o=0;o<d3;o++){
          double cr=U3r[c][o], ci=U3i[c][o];
          sr += cr*A2r[a][b][o]-ci*A2i[a][b][o];
          si += cr*A2i[a][b][o]+ci*A2r[a][b][o];
        }
        double val = par ? si : sr;
        ws[WS_CG + (l1*l1+a)*81 + (l2*l2+b)*9 + (l3*l3+c)] = (float)val;
      }
    }
  }
  __syncthreads();

  for (int i=tid;i<729;i+=256){
    int a=i/81, b=(i/9)%9, c=i%9;
    ws[WS_WCG+i] = ws[WS_CG+i] * wpath[lof9[a]*9 + lof9[b]*3 + lof9[c]];
  }
  for (int i=tid;i<81;i+=256){
    int e=i/9, l=i%9; int a=e/3, b=e%3;
    ws[WS_CG112+i] = ws[WS_CG + (1+a)*81 + (1+b)*9 + l];
  }
}

// ---------------- main streaming kernel ----------------
// Per wave: 8 n-values = 144 contiguous rows of sf = 9 tiles of 16x128.
// Double-buffered async staging (GLOBAL_LOAD_ASYNC_TO_LDS_B128, ASYNCcnt),
// V_WMMA_F32_16X16X4_F32 K-loop, B (128x16 collapsed weights) in VGPRs.
// Waves are fully independent: intra-wave LDS ordering only (DS is in-order
// per wave; fences keep the compiler honest).

#define STG_PITCH    132                 // 128 + 4 dword pad: 16B rows, conflict-free DS
#define STAGE_FLOATS (16*STG_PITCH)      // 2112
#define LDS_PER_WAVE (2*STAGE_FLOATS + 16*18 + 2*8*18)  // 4800 floats

__device__ __forceinline__ void wave_lds_sync(){
  __builtin_amdgcn_fence(__ATOMIC_ACQ_REL, "wavefront");
  __builtin_amdgcn_wave_barrier();
  __builtin_amdgcn_fence(__ATOMIC_ACQ_REL, "wavefront");
}

__device__ __forceinline__ void async_load_b128(unsigned lds_off, const float* gaddr){
  // VDST = per-lane LDS byte address (workgroup-relative), VADDR = 64-bit global addr
  asm volatile("global_load_async_to_lds_b128 %0, %1, off"
               :: "v"(lds_off), "v"(gaddr) : "memory");
}
#define WAIT_ASYNC_LE_16() asm volatile("s_wait_asynccnt 16" ::: "memory")
#define WAIT_ASYNC_LE_0()  asm volatile("s_wait_asynccnt 0"  ::: "memory")

__global__ __launch_bounds__(WG_SIZE)
void pptp_main(const float* __restrict__ sf, const float* __restrict__ ws,
               float* __restrict__ out)
{
  __shared__ float lds[WAVES_PER_WG*LDS_PER_WAVE + 729 + 81 + 2];
  const int tid  = threadIdx.x;
  const int wave = tid >> 5;
  const int lane = tid & 31;

  float* stage0 = lds + wave*LDS_PER_WAVE;            // 16 x 132
  float* stage1 = stage0 + STAGE_FLOATS;              // 16 x 132
  float* Cd     = stage0 + 2*STAGE_FLOATS;            // 16 x 18 (padded C dump)
  float* xbuf   = Cd + 16*18;                         // [t(2)][n(8)][rem(18)]
  float* wcgs   = lds + WAVES_PER_WG*LDS_PER_WAVE;    // 729
  float* cg112s = wcgs + 729;                         // 81
  float* cbs    = cg112s + 81;                        // 2

  // cooperative load of small tables (ws[WS_WCG..] is a contiguous 812-float blob)
  for (int i=tid; i<812; i+=WG_SIZE) lds[WAVES_PER_WG*LDS_PER_WAVE + i] = ws[WS_WCG + i];

  // B fragments (loop-invariant): lane<16 holds B[k0],B[k0+1] col=lane; lane>=16 holds B[k0+2],B[k0+3]
  v2f bfrag[32];
  #pragma unroll
  for (int s=0;s<32;++s){
    int k0  = 4*s + ((lane<16)?0:2);
    int col = lane & 15;
    bfrag[s].x = ws[WS_U + (k0+0)*16 + col];
    bfrag[s].y = ws[WS_U + (k0+1)*16 + col];
  }
  __syncthreads();   // tables ready (only inter-wave sync in the kernel)

  const int wblock = blockIdx.x * WAVES_PER_WG + wave;     // 0..8191
  const int n0 = wblock * N_PER_WAVE;
  const float* base = sf + (size_t)n0 * ROWS_PER_N * F;

  // per-lane LDS byte addresses of the two staging buffers (+ lane's 16B column)
  const unsigned sOff0 = (unsigned)(size_t)stage0 + (unsigned)lane*16u;
  const unsigned sOff1 = (unsigned)(size_t)stage1 + (unsigned)lane*16u;

  // prologue: async-stage tile 0 into buffer 0
  {
    const float* tb = base + lane*4;
    #pragma unroll
    for (int rr=0; rr<16; ++rr)
      async_load_b128(sOff0 + rr*(STG_PITCH*4), tb + rr*F);
  }

  for (int tile=0; tile<TILES_PER_WAVE; ++tile){
    const int cur = tile & 1;
    const unsigned curOff = cur ? sOff1 : sOff0;
    float* curStage = cur ? stage1 : stage0;

    // issue async stage of tile+1 into the other buffer (read 2 tiles ago -> drained)
    if (tile+1 < TILES_PER_WAVE){
      const unsigned nxtOff = cur ? sOff0 : sOff1;
      const float* tb = base + (size_t)(tile+1)*16*F + lane*4;
      #pragma unroll
      for (int rr=0; rr<16; ++rr)
        async_load_b128(nxtOff + rr*(STG_PITCH*4), tb + rr*F);
      WAIT_ASYNC_LE_16();   // in-order retirement: current tile's 16 ops are done
    } else {
      WAIT_ASYNC_LE_0();
    }
    wave_lds_sync();

    // K-loop: 32 x V_WMMA_F32_16X16X4_F32
    v8f acc = {};
    const int r    = lane & 15;
    const int kadj = (lane<16) ? 0 : 2;
    const float* arow = curStage + r*STG_PITCH + kadj;
    #pragma unroll
    for (int s=0;s<32;++s){
      v2f a = *(const v2f*)(arow + 4*s);
      acc = __builtin_amdgcn_wmma_f32_16x16x4_f32(false, a, false, bfrag[s],
                                                  (short)0, acc, false, false);
    }

    // dump C (16x16) to LDS with pitch 18 (conflict-free)
    const int nsel = lane & 15;
    const int mhi  = (lane>>4)*8;
    #pragma unroll
    for (int v=0; v<8; ++v) Cd[(v+mhi)*18 + nsel] = acc[v];
    wave_lds_sync();

    // extraction: lane<16 pulls x1 for row m=lane, lane>=16 pulls x2
    {
      int m    = lane & 15;
      int tsel = lane >> 4;
      int q    = tile*16 + m;          // flat row within 8-n block
      int nl   = q/18, rem = q - nl*18;
      int p    = rem/9, lm = rem - p*9;
      int l    = (lm==0)?0 : ((lm<4)?1:2);
      float val = Cd[m*18 + (tsel*6 + p*3 + l)];
      if (rem==0) val += cbs[tsel];
      xbuf[(tsel*8 + nl)*18 + rem] = val;
    }
    wave_lds_sync();
  }

  // final tiny bilinear CG contraction: lanes 0..7, one n each
  if (lane < N_PER_WAVE){
    const int nl = lane;
    float x1v[18], x2v[18];
    #pragma unroll
    for (int i=0;i<18;++i){ x1v[i]=xbuf[nl*18+i]; x2v[i]=xbuf[(8+nl)*18+i]; }
    float v[9];
    #pragma unroll
    for (int c=0;c<9;++c) v[c]=0.f;
    for (int p=0;p<2;++p)
      for (int a=0;a<9;++a){
        float xa = x1v[p*9+a];
        for (int b=0;b<9;++b){
          float f = xa * x2v[p*9+b];
          #pragma unroll
          for (int c=0;c<9;++c) v[c] += f * wcgs[(a*9+b)*9 + c];
        }
      }
    float* op = out + (size_t)(n0+nl)*9;
    #pragma unroll
    for (int e=0;e<9;++e){
      float s=0.f;
      #pragma unroll
      for (int l=0;l<9;++l) s += v[l]*cg112s[e*9+l];
      op[e] = s;
    }
  }
}

// ---------------- launch ----------------

extern "C" void kernel_launch(void* const* d_in, const int* in_sizes, int n_in,
                              void* d_out, int out_size, void* d_ws, size_t ws_size,
                              hipStream_t stream) {
  const float* sf    = (const float*)d_in[0];
  const float* W3    = (const float*)d_in[1];
  const float* b3    = (const float*)d_in[2];
  const float* Wt1   = (const float*)d_in[3];
  const float* bt1   = (const float*)d_in[4];
  const float* Wt2   = (const float*)d_in[5];
  const float* bt2   = (const float*)d_in[6];
  const float* wpath = (const float*)d_in[7];
  float* wsf = (float*)d_ws;
  float* out = (float*)d_out;

  hipLaunchKernelGGL(pptp_precompute, dim3(1), dim3(256), 0, stream,
                     W3, b3, Wt1, bt1, Wt2, bt2, wpath, wsf);

  const int num_wblocks = N_TOT / N_PER_WAVE;          // 8192
  const int num_wgs = num_wblocks / WAVES_PER_WG;      // 4096
  hipLaunchKernelGGL(pptp_main, dim3(num_wgs), dim3(WG_SIZE), 0, stream,
                     sf, wsf, out);
}